// GraphNN_82136954568750
// MI455X (gfx1250) — compile-verified
//
#include <hip/hip_runtime.h>
#include <hip/hip_bf16.h>

typedef __attribute__((ext_vector_type(2))) float v2f;
typedef __attribute__((ext_vector_type(8))) float v8f;

#define DIM 64

// ---------------------------------------------------------------- fill zeros
__global__ void fill_zero_kernel(float* __restrict__ p, long long n) {
    long long i = (long long)blockIdx.x * blockDim.x + threadIdx.x;
    if (i < n) p[i] = 0.0f;
}

// ------------------------------------------------- deg/cnt accumulation (once)
__global__ void deg_cnt_kernel(const int* __restrict__ col,
                               const float* __restrict__ ew,
                               float* __restrict__ deg,
                               float* __restrict__ cnt, int E) {
    int e = blockIdx.x * blockDim.x + threadIdx.x;
    if (e >= E) return;
    int c = col[e];
    atomicAdd(&deg[c], ew[e]);
    atomicAdd(&cnt[c], 1.0f);
}

// ------------------------------------- dis = rsqrt(deg) | cnt_inv = 1/max(cnt,1)
__global__ void normalize_kernel(float* __restrict__ deg,
                                 float* __restrict__ cnt, int N) {
    int i = blockIdx.x * blockDim.x + threadIdx.x;
    if (i >= N) return;
    float d = deg[i];
    deg[i] = (d > 0.0f) ? rsqrtf(d) : 0.0f;   // deg now holds dis
    float c = cnt[i];
    cnt[i] = 1.0f / fmaxf(c, 1.0f);           // cnt now holds 1/max(cnt,1)
}

// ----------------------------------------- w[e] = dis[row]*ew*dis[col] (once)
__global__ void wnorm_kernel(const int* __restrict__ row,
                             const int* __restrict__ col,
                             const float* __restrict__ ew,
                             const float* __restrict__ dis,
                             float* __restrict__ wn, int E) {
    int e = blockIdx.x * blockDim.x + threadIdx.x;
    if (e >= E) return;
    wn[e] = dis[row[e]] * ew[e] * dis[col[e]];
}

// ---------------------- scatter: agg[col] += w * x[row]; 16 threads per edge
__global__ void scatter_kernel(const float* __restrict__ x,
                               const int* __restrict__ row,
                               const int* __restrict__ col,
                               const float* __restrict__ wn,
                               float* __restrict__ agg, int E) {
    long long tid = (long long)blockIdx.x * blockDim.x + threadIdx.x;
    int e = (int)(tid >> 4);
    if (e >= E) return;
    int d = ((int)tid & 15) << 2;             // 4 dims per thread
    int r = row[e], c = col[e];
    float w = wn[e];
    const float4 xv = *reinterpret_cast<const float4*>(x + (size_t)r * DIM + d);
    float* dst = agg + (size_t)c * DIM + d;
    atomicAdd(dst + 0, w * xv.x);
    atomicAdd(dst + 1, w * xv.y);
    atomicAdd(dst + 2, w * xv.z);
    atomicAdd(dst + 3, w * xv.w);
}

// --------------------------------------------------------------- WMMA GEMM
// out[16n x 64] = (agg*cnt_inv) @ Wl^T + x @ Wr^T + bl  (optional ReLU)
// One wave32 per 16-node tile, V_WMMA_F32_16X16X4_F32, fp32 exact.
// f32 A 16x4 layout: lane m=lane&15, K = 2*(lane>>4)+v  (v = vgpr 0/1)
// f32 B 4x16 layout: lane n=lane&15, K = 2*(lane>>4)+v
// f32 C/D 16x16:     lane n=lane&15, row = r + 8*(lane>>4) for vgpr r
__global__ void gcn_gemm_wmma_kernel(const float* __restrict__ agg,
                                     const float* __restrict__ cnt_inv,
                                     const float* __restrict__ xin,
                                     const float* __restrict__ Wl,
                                     const float* __restrict__ bl,
                                     const float* __restrict__ Wr,
                                     float* __restrict__ out,
                                     int N, int relu) {
    int wave = (int)(((long long)blockIdx.x * blockDim.x + threadIdx.x) >> 5);
    int lane = threadIdx.x & 31;
    int node0 = wave * 16;
    if (node0 >= N) return;                   // wave-uniform: EXEC all-1s for WMMA

    const int m = lane & 15;                  // A row / D column within tile
    const int h = lane >> 4;                  // half-wave K selector
    const int node = node0 + m;
    const float ci = cnt_inv[node];

    // Preload this lane's A data: 16 k-steps x (K=4kk+2h, +1)
    v2f aA[16], aX[16];
    const float* arow = agg + (size_t)node * DIM;
    const float* xrow = xin + (size_t)node * DIM;
#pragma unroll
    for (int kk = 0; kk < 16; ++kk) {
        int k = 4 * kk + 2 * h;
        float2 t = *reinterpret_cast<const float2*>(arow + k);
        float2 u = *reinterpret_cast<const float2*>(xrow + k);
        aA[kk].x = t.x * ci; aA[kk].y = t.y * ci;   // mean-normalize the A rows
        aX[kk].x = u.x;      aX[kk].y = u.y;
    }

#pragma unroll
    for (int j = 0; j < 4; ++j) {             // 4 output column tiles of 16
        const int n = 16 * j + m;             // this lane's output column
        const float bias = bl[n];
        v8f acc = {bias, bias, bias, bias, bias, bias, bias, bias};
        // B[k][n] = W[n][k] since out = A @ W^T ; W is [64 x 64] row-major
        const float* wlc = Wl + (size_t)n * DIM;
        const float* wrc = Wr + (size_t)n * DIM;
#pragma unroll
        for (int kk = 0; kk < 16; ++kk) {
            int k = 4 * kk + 2 * h;
            float2 tl = *reinterpret_cast<const float2*>(wlc + k);
            float2 tr = *reinterpret_cast<const float2*>(wrc + k);
            v2f bL; bL.x = tl.x; bL.y = tl.y;
            v2f bR; bR.x = tr.x; bR.y = tr.y;
            acc = __builtin_amdgcn_wmma_f32_16x16x4_f32(
                false, aA[kk], false, bL, (short)0, acc, false, false);
            acc = __builtin_amdgcn_wmma_f32_16x16x4_f32(
                false, aX[kk], false, bR, (short)0, acc, false, false);
        }
        if (relu) {
#pragma unroll
            for (int r = 0; r < 8; ++r) acc[r] = fmaxf(acc[r], 0.0f);
        }
#pragma unroll
        for (int r = 0; r < 8; ++r) {
            int mrow = r + 8 * h;
            out[(size_t)(node0 + mrow) * DIM + n] = acc[r];
        }
    }
}

// ----------------------------------------------------------------- launcher
static void run_layer(const float* xin, float* hout,
                      const float* Wl, const float* bl, const float* Wr,
                      const int* row, const int* col,
                      const float* wn, const float* cnt_inv, float* agg,
                      int N, int E, int relu, hipStream_t stream) {
    long long aggN = (long long)N * DIM;
    fill_zero_kernel<<<(int)((aggN + 255) / 256), 256, 0, stream>>>(agg, aggN);
    long long sth = (long long)E * 16;
    scatter_kernel<<<(int)((sth + 255) / 256), 256, 0, stream>>>(
        xin, row, col, wn, agg, E);
    int tiles = (N + 15) / 16;                 // waves
    int blocks = (tiles + 7) / 8;              // 8 waves (256 thr) per block
    gcn_gemm_wmma_kernel<<<blocks, 256, 0, stream>>>(
        agg, cnt_inv, xin, Wl, bl, Wr, hout, N, relu);
}

extern "C" void kernel_launch(void* const* d_in, const int* in_sizes, int n_in,
                              void* d_out, int out_size, void* d_ws, size_t ws_size,
                              hipStream_t stream) {
    const float* x   = (const float*)d_in[0];
    const int*   ei  = (const int*)d_in[1];     // [2, E]
    const float* ew  = (const float*)d_in[2];
    const float* Wl0 = (const float*)d_in[3];
    const float* bl0 = (const float*)d_in[4];
    const float* Wr0 = (const float*)d_in[5];
    const float* Wl1 = (const float*)d_in[6];
    const float* bl1 = (const float*)d_in[7];
    const float* Wr1 = (const float*)d_in[8];
    const float* Wl2 = (const float*)d_in[9];
    const float* bl2 = (const float*)d_in[10];
    const float* Wr2 = (const float*)d_in[11];

    const int E = in_sizes[2];
    const int N = in_sizes[0] / DIM;
    const int* row = ei;
    const int* col = ei + E;

    // workspace carve-up (all sizes 16B aligned for N=100000, E=1000000)
    char* ws = (char*)d_ws;
    float* deg = (float*)ws;                 ws += (size_t)N * 4;   // -> dis
    float* cnt = (float*)ws;                 ws += (size_t)N * 4;   // -> 1/max(cnt,1)
    float* wn  = (float*)ws;                 ws += (size_t)E * 4;
    float* agg = (float*)ws;                 ws += (size_t)N * DIM * 4;
    float* h0  = (float*)ws;                 ws += (size_t)N * DIM * 4;
    float* h1  = (float*)ws;

    // edge-weight normalization is layer-invariant: compute once
    long long twoN = 2LL * N;                // deg and cnt are contiguous
    fill_zero_kernel<<<(int)((twoN + 255) / 256), 256, 0, stream>>>(deg, twoN);
    deg_cnt_kernel<<<(E + 255) / 256, 256, 0, stream>>>(col, ew, deg, cnt, E);
    normalize_kernel<<<(N + 255) / 256, 256, 0, stream>>>(deg, cnt, N);
    wnorm_kernel<<<(E + 255) / 256, 256, 0, stream>>>(row, col, ew, deg, wn, E);

    run_layer(x,  h0, Wl0, bl0, Wr0, row, col, wn, cnt, agg, N, E, 1, stream);
    run_layer(h0, h1, Wl1, bl1, Wr1, row, col, wn, cnt, agg, N, E, 1, stream);
    run_layer(h1, (float*)d_out, Wl2, bl2, Wr2, row, col, wn, cnt, agg, N, E, 0, stream);
}